// DummyGAT_1726576853266
// MI455X (gfx1250) — compile-verified
//
#include <hip/hip_runtime.h>

typedef __attribute__((ext_vector_type(2))) float v2f;
typedef __attribute__((ext_vector_type(8))) float v8f;

#define HID 64
#define NEG_SLOPE 0.2f
// enc(-inf): ~0xFF800000
#define ENC_NEG_INF 0x007FFFFFu

// Monotone order-preserving float->uint encoding (for atomic max on floats).
__device__ __forceinline__ unsigned enc_f32(float f) {
    unsigned u = __float_as_uint(f);
    return (u & 0x80000000u) ? ~u : (u | 0x80000000u);
}
__device__ __forceinline__ float dec_f32(unsigned u) {
    u = (u & 0x80000000u) ? (u & 0x7FFFFFFFu) : ~u;
    return __uint_as_float(u);
}

// ---------------------------------------------------------------------------
// h = x @ W via V_WMMA_F32_16X16X4_F32. One 16x16 output tile per wave,
// 4 waves per block cover HID=64 columns; grid.x tiles the M dimension.
// ---------------------------------------------------------------------------
__global__ void gat_gemm_h(const float* __restrict__ x, const float* __restrict__ W,
                           float* __restrict__ h, int N) {
    const int lane = threadIdx.x & 31;
    const int wave = threadIdx.x >> 5;      // 0..3 -> column tile
    const int m0   = blockIdx.x * 16;
    const int n0   = wave * 16;
    const int half = lane >> 4;             // 0 or 1 (wave32 half)
    const int lr   = lane & 15;
    if (m0 + 16 > N) return;                // full tiles only (tail kernel handles rest)

    v8f acc = {};
    const float* arow = x + (size_t)(m0 + lr) * HID;
#pragma unroll
    for (int k = 0; k < HID; k += 4) {
        const int ka = k + 2 * half;
        v2f a, b;
        a.x = arow[ka];
        a.y = arow[ka + 1];
        b.x = W[(size_t)ka * HID + n0 + lr];
        b.y = W[(size_t)(ka + 1) * HID + n0 + lr];
        // 8-arg probe-confirmed form: (neg_a, A, neg_b, B, c_mod, C, reuse_a, reuse_b)
        acc = __builtin_amdgcn_wmma_f32_16x16x4_f32(false, a, false, b,
                                                    (short)0, acc, false, false);
    }
#pragma unroll
    for (int v = 0; v < 8; ++v)
        h[(size_t)(m0 + v + 8 * half) * HID + n0 + lr] = acc[v];
}

// Scalar fallback for a non-multiple-of-16 row tail (N=50000 -> never launched).
__global__ void gat_gemm_tail(const float* __restrict__ x, const float* __restrict__ W,
                              float* __restrict__ h, int rowBase, int N) {
    int t = blockIdx.x * blockDim.x + threadIdx.x;
    int row = rowBase + t / HID;
    int col = t % HID;
    if (row >= N) return;
    float acc = 0.f;
#pragma unroll
    for (int k = 0; k < HID; ++k)
        acc += x[(size_t)row * HID + k] * W[(size_t)k * HID + col];
    h[(size_t)row * HID + col] = acc;
}

// a_src[n] = h[n,:].att_src ; a_dst[n] = h[n,:].att_dst  (one node per thread)
__global__ void gat_attn_scores(const float* __restrict__ h,
                                const float* __restrict__ att_src,
                                const float* __restrict__ att_dst,
                                float* __restrict__ a_src, float* __restrict__ a_dst,
                                int N) {
    int n = blockIdx.x * blockDim.x + threadIdx.x;
    if (n >= N) return;
    const float4* hp = (const float4*)(h + (size_t)n * HID);
    const float4* as = (const float4*)att_src;
    const float4* ad = (const float4*)att_dst;
    float ssum = 0.f, dsum = 0.f;
#pragma unroll
    for (int k = 0; k < HID / 4; ++k) {
        float4 hv = hp[k], sv = as[k], dv = ad[k];
        ssum += hv.x * sv.x + hv.y * sv.y + hv.z * sv.z + hv.w * sv.w;
        dsum += hv.x * dv.x + hv.y * dv.y + hv.z * dv.z + hv.w * dv.w;
    }
    a_src[n] = ssum;
    a_dst[n] = dsum;
}

// Zero/seed all accumulators (d_ws is poisoned by the harness).
__global__ void gat_init(float* __restrict__ agg, float* __restrict__ denom,
                         unsigned* __restrict__ emax_enc, float* __restrict__ colsum,
                         int N) {
    size_t i = (size_t)blockIdx.x * blockDim.x + threadIdx.x;
    if (i < (size_t)N * HID) agg[i] = 0.f;
    if (i < (size_t)N) { denom[i] = 0.f; emax_enc[i] = ENC_NEG_INF; }
    if (i < HID) colsum[i] = 0.f;
}

// Pass 1 over E+N messages: e = leaky_relu(a_src[s]+a_dst[d]); segment max.
__global__ void gat_edge_max(const int* __restrict__ esrc, const int* __restrict__ edst,
                             const float* __restrict__ a_src, const float* __restrict__ a_dst,
                             float* __restrict__ ebuf, unsigned* __restrict__ emax_enc,
                             int E, int N) {
    int i = blockIdx.x * blockDim.x + threadIdx.x;
    if (i >= E + N) return;
    int s, d;
    if (i < E) { s = esrc[i]; d = edst[i]; } else { s = d = i - E; }
    float e = a_src[s] + a_dst[d];
    e = (e > 0.f) ? e : NEG_SLOPE * e;
    ebuf[i] = e;
    atomicMax(&emax_enc[d], enc_f32(e));
}

// Pass 2: ex = exp(e - emax[d]) (in place); segment sum of denominators.
__global__ void gat_edge_exp(const int* __restrict__ edst,
                             const unsigned* __restrict__ emax_enc,
                             float* __restrict__ ebuf, float* __restrict__ denom,
                             int E, int N) {
    int i = blockIdx.x * blockDim.x + threadIdx.x;
    if (i >= E + N) return;
    int d = (i < E) ? edst[i] : (i - E);
    float ex = __expf(ebuf[i] - dec_f32(emax_enc[d]));
    ebuf[i] = ex;
    atomicAdd(&denom[d], ex);
}

// Pass 3: agg[d,:] += (ex/denom[d]) * h[s,:]. 64 threads per message (2 waves),
// coalesced 64-float gather + 64 L2-resident float atomics.
__global__ void gat_edge_agg(const int* __restrict__ esrc, const int* __restrict__ edst,
                             const float* __restrict__ ebuf, const float* __restrict__ denom,
                             const float* __restrict__ h, float* __restrict__ agg,
                             int E, int N) {
    long long gid = (long long)blockIdx.x * blockDim.x + threadIdx.x;
    long long i = gid >> 6;                  // message id
    int k = (int)(gid & 63);                 // feature id
    if (i >= (long long)E + N) return;
    int s, d;
    if (i < E) { s = esrc[i]; d = edst[i]; } else { s = d = (int)(i - E); }
    float alpha = ebuf[i] / denom[d];
    atomicAdd(&agg[(size_t)d * HID + k], alpha * h[(size_t)s * HID + k]);
}

// Column sums of relu(agg + bias): mean commutes with the final affine map.
#define CS_ROWS 128
__global__ void gat_colsum(const float* __restrict__ agg, const float* __restrict__ bias,
                           float* __restrict__ colsum, int N) {
    __shared__ float sm[256];
    int col  = threadIdx.x & 63;
    int rofs = threadIdx.x >> 6;             // 0..3
    int r0   = blockIdx.x * CS_ROWS;
    int rEnd = r0 + CS_ROWS; if (rEnd > N) rEnd = N;
    float b = bias[col];
    float acc = 0.f;
    for (int r = r0 + rofs; r < rEnd; r += 4) {
        float v = agg[(size_t)r * HID + col] + b;
        acc += (v > 0.f) ? v : 0.f;
    }
    sm[threadIdx.x] = acc;
    __syncthreads();
    if (threadIdx.x < 64) {
        float t = sm[threadIdx.x] + sm[threadIdx.x + 64] +
                  sm[threadIdx.x + 128] + sm[threadIdx.x + 192];
        atomicAdd(&colsum[col], t);
    }
}

// out[1,64] = (colsum/N) @ lin_w + lin_b
__global__ void gat_final(const float* __restrict__ colsum, const float* __restrict__ lin_w,
                          const float* __restrict__ lin_b, float* __restrict__ out, int N) {
    int o = threadIdx.x;
    if (o >= HID) return;
    const float invN = 1.0f / (float)N;
    float acc = 0.f;
#pragma unroll
    for (int k = 0; k < HID; ++k)
        acc += (colsum[k] * invN) * lin_w[(size_t)k * HID + o];
    out[o] = acc + lin_b[o];
}

extern "C" void kernel_launch(void* const* d_in, const int* in_sizes, int n_in,
                              void* d_out, int out_size, void* d_ws, size_t ws_size,
                              hipStream_t stream) {
    const float* x       = (const float*)d_in[0];
    const int*   eidx    = (const int*)d_in[1];
    const float* W       = (const float*)d_in[2];
    const float* att_src = (const float*)d_in[3];
    const float* att_dst = (const float*)d_in[4];
    const float* bias    = (const float*)d_in[5];
    const float* lin_w   = (const float*)d_in[6];
    const float* lin_b   = (const float*)d_in[7];
    float* out = (float*)d_out;

    const int N = in_sizes[0] / HID;
    const int E = in_sizes[1] / 2;
    const int* esrc = eidx;
    const int* edst = eidx + E;

    // Workspace layout (~30 MB; everything L2-resident on MI455X)
    float*    h        = (float*)d_ws;                      // N*64
    float*    a_src    = h + (size_t)N * HID;               // N
    float*    a_dst    = a_src + N;                         // N
    float*    denom    = a_dst + N;                         // N
    unsigned* emax_enc = (unsigned*)(denom + N);            // N
    float*    ebuf     = (float*)(emax_enc + N);            // E+N
    float*    agg      = ebuf + ((size_t)E + N);            // N*64
    float*    colsum   = agg + (size_t)N * HID;             // 64

    const int M = E + N;                                    // messages incl. self-loops

    // init accumulators
    {
        size_t total = (size_t)N * HID;
        int blocks = (int)((total + 255) / 256);
        gat_init<<<blocks, 256, 0, stream>>>(agg, denom, emax_enc, colsum, N);
    }
    // h = x @ W  (WMMA fp32)
    {
        int mtiles = N / 16;
        if (mtiles > 0)
            gat_gemm_h<<<mtiles, 128, 0, stream>>>(x, W, h, N);
        int rem = N - mtiles * 16;
        if (rem > 0) {
            int threads = rem * HID;
            gat_gemm_tail<<<(threads + 255) / 256, 256, 0, stream>>>(x, W, h, mtiles * 16, N);
        }
    }
    // per-node attention scores
    gat_attn_scores<<<(N + 255) / 256, 256, 0, stream>>>(h, att_src, att_dst, a_src, a_dst, N);
    // segment softmax over incoming edges
    gat_edge_max<<<(M + 255) / 256, 256, 0, stream>>>(esrc, edst, a_src, a_dst, ebuf, emax_enc, E, N);
    gat_edge_exp<<<(M + 255) / 256, 256, 0, stream>>>(edst, emax_enc, ebuf, denom, E, N);
    // weighted aggregation
    {
        long long threads = (long long)M * HID;
        int blocks = (int)((threads + 255) / 256);
        gat_edge_agg<<<blocks, 256, 0, stream>>>(esrc, edst, ebuf, denom, h, agg, E, N);
    }
    // relu + bias + column mean, then final 64x64 matvec
    gat_colsum<<<(N + CS_ROWS - 1) / CS_ROWS, 256, 0, stream>>>(agg, bias, colsum, N);
    gat_final<<<1, 64, 0, stream>>>(colsum, lin_w, lin_b, out, N);
}